// PositionEncoder_50560355009095
// MI455X (gfx1250) — compile-verified
//
#include <hip/hip_runtime.h>

typedef float v2f __attribute__((ext_vector_type(2)));
typedef float v8f __attribute__((ext_vector_type(8)));

#define GCN_NFEAT 128
#define GCN_HID   64

// ---------------------------------------------------------------------------
// Degree: deg[i] = 1 (self loop) + #incoming edges; then deg <- rsqrt(deg)
// ---------------------------------------------------------------------------
__global__ void k_deg_init(float* __restrict__ deg, int n) {
    int i = blockIdx.x * blockDim.x + threadIdx.x;
    if (i < n) deg[i] = 1.0f;
}

__global__ void k_deg_scatter(const int* __restrict__ dst, float* __restrict__ deg, int E) {
    int e = blockIdx.x * blockDim.x + threadIdx.x;
    if (e < E) unsafeAtomicAdd(&deg[dst[e]], 1.0f);
}

__global__ void k_rsqrt_inplace(float* __restrict__ deg, int n) {
    int i = blockIdx.x * blockDim.x + threadIdx.x;
    if (i < n) deg[i] = rsqrtf(deg[i]);
}

// ---------------------------------------------------------------------------
// C[M x 64] = A[M x K] @ B[K x 64] (+ optional bias[64])  via V_WMMA_F32_16X16X4_F32
// Block = 128 threads (4 waves). Wave w computes the 16x16 tile at
// rows [blockIdx.x*16, +16), cols [w*16, +16).  M must be a multiple of 16.
//
// Per-lane VGPR layout for the f32 WMMA (wave32):
//   half = lane>>4, l = lane&15
//   A (16x4):  a[j] = A[row=l][k0 + 2*half + j]        (contiguous pair -> float2 load)
//   B (4x16):  b[j] = B[k0 + 2*half + j][col=l]
//   C (16x16): acc[r] = C[row = r + 8*half][col = l]
// ---------------------------------------------------------------------------
template <int K>
__global__ void k_gemm64_wmma(const float* __restrict__ A,
                              const float* __restrict__ B,
                              const float* __restrict__ bias,
                              float* __restrict__ C) {
    const int wave = threadIdx.x >> 5;
    const int lane = threadIdx.x & 31;
    const int half = lane >> 4;
    const int l    = lane & 15;
    const int row0 = blockIdx.x * 16;
    const int col  = wave * 16 + l;

    const float binit = bias ? bias[col] : 0.0f;
    v8f acc;
#pragma unroll
    for (int r = 0; r < 8; ++r) acc[r] = binit;

    const float* __restrict__ arow = A + (size_t)(row0 + l) * K;

#pragma unroll
    for (int k0 = 0; k0 < K; k0 += 4) {
        const int k = k0 + 2 * half;
        v2f a = *(const v2f*)(arow + k);
        v2f b;
        b.x = B[(size_t)k * 64 + col];
        b.y = B[(size_t)(k + 1) * 64 + col];
        // 8 args: (neg_a, A, neg_b, B, c_mod, C, reuse_a, reuse_b)
        acc = __builtin_amdgcn_wmma_f32_16x16x4_f32(false, a, false, b,
                                                    (short)0, acc, false, false);
    }

#pragma unroll
    for (int r = 0; r < 8; ++r)
        C[(size_t)(row0 + r + 8 * half) * 64 + col] = acc[r];
}

// ---------------------------------------------------------------------------
// agg[i][c] = hw[i][c] * dis[i]^2 + bias[c]   (self-loop term + bias)
// Fully (re)writes `out` each call -> also clears the harness poison.
// ---------------------------------------------------------------------------
__global__ void k_self_bias(const float* __restrict__ hw,
                            const float* __restrict__ dis,
                            const float* __restrict__ bias,
                            float* __restrict__ out, int total) {
    int i = blockIdx.x * blockDim.x + threadIdx.x;
    if (i < total) {
        int node = i >> 6;
        int c    = i & 63;
        float d  = dis[node];
        out[i] = hw[i] * d * d + bias[c];
    }
}

// ---------------------------------------------------------------------------
// Edge scatter: one edge per wave32; lane handles columns {2*lane, 2*lane+1}.
// out[dst] += hw[src] * dis[src]*dis[dst]   via global_atomic_add_f32.
// ---------------------------------------------------------------------------
__global__ void k_edge_scatter(const int* __restrict__ src,
                               const int* __restrict__ dst,
                               const float* __restrict__ hw,
                               const float* __restrict__ dis,
                               float* __restrict__ out, int E) {
    int gw   = (int)((blockIdx.x * (unsigned)blockDim.x + threadIdx.x) >> 5);
    int lane = threadIdx.x & 31;
    if (gw >= E) return;
    int s = src[gw];
    int d = dst[gw];
    float coef = dis[s] * dis[d];
    v2f m = *(const v2f*)(hw + (size_t)s * 64 + 2 * lane);
    float* p = out + (size_t)d * 64 + 2 * lane;
    unsafeAtomicAdd(p,     m.x * coef);
    unsafeAtomicAdd(p + 1, m.y * coef);
}

__global__ void k_relu_copy(const float* __restrict__ in, float* __restrict__ out, int total) {
    int i = blockIdx.x * blockDim.x + threadIdx.x;
    if (i < total) out[i] = fmaxf(in[i], 0.0f);
}

// ---------------------------------------------------------------------------
// Orchestration
// inputs: x[N,128] f32 | edge_index[2,E] int | W_emb[128,64] | b_emb[64]
//         Ws[3,64,64] | bs[3,64]
// output: h[N,64] f32
// ---------------------------------------------------------------------------
extern "C" void kernel_launch(void* const* d_in, const int* in_sizes, int n_in,
                              void* d_out, int out_size, void* d_ws, size_t ws_size,
                              hipStream_t stream) {
    const float* x     = (const float*)d_in[0];
    const int*   edge  = (const int*)d_in[1];
    const float* W_emb = (const float*)d_in[2];
    const float* b_emb = (const float*)d_in[3];
    const float* Ws    = (const float*)d_in[4];
    const float* bs    = (const float*)d_in[5];
    float* out = (float*)d_out;

    const int N = in_sizes[0] / GCN_NFEAT;   // 100000
    const int E = in_sizes[1] / 2;           // 1600000
    const int* srcIdx = edge;
    const int* dstIdx = edge + E;

    float* h   = (float*)d_ws;               // N*64 floats
    float* hw  = h  + (size_t)N * GCN_HID;   // N*64 floats
    float* dis = hw + (size_t)N * GCN_HID;   // N floats (deg -> rsqrt in place)

    const int total = N * GCN_HID;
    const int TPB = 256;

    // Degree with self-loops, then rsqrt (shared by all layers)
    k_deg_init<<<(N + TPB - 1) / TPB, TPB, 0, stream>>>(dis, N);
    k_deg_scatter<<<(E + TPB - 1) / TPB, TPB, 0, stream>>>(dstIdx, dis, E);
    k_rsqrt_inplace<<<(N + TPB - 1) / TPB, TPB, 0, stream>>>(dis, N);

    // Embedding: h = x @ W_emb + b_emb   (N % 16 == 0)
    k_gemm64_wmma<GCN_NFEAT><<<N / 16, 128, 0, stream>>>(x, W_emb, b_emb, h);

    for (int layer = 0; layer < 3; ++layer) {
        const float* Wl = Ws + (size_t)layer * GCN_HID * GCN_HID;
        const float* bl = bs + (size_t)layer * GCN_HID;

        // hw = h @ Wl
        k_gemm64_wmma<GCN_HID><<<N / 16, 128, 0, stream>>>(h, Wl, nullptr, hw);

        // out = hw * dis^2 + bl   (self-loop + bias; full overwrite of d_out)
        k_self_bias<<<(total + TPB - 1) / TPB, TPB, 0, stream>>>(hw, dis, bl, out, total);

        // out[dst] += hw[src] * dis[src]*dis[dst]   (one edge per wave32)
        {
            long long threads = (long long)E * 32;
            int blocks = (int)((threads + TPB - 1) / TPB);
            k_edge_scatter<<<blocks, TPB, 0, stream>>>(srcIdx, dstIdx, hw, dis, out, E);
        }

        // relu -> next layer input (or finalize in place on last layer)
        if (layer < 2)
            k_relu_copy<<<(total + TPB - 1) / TPB, TPB, 0, stream>>>(out, h, total);
        else
            k_relu_copy<<<(total + TPB - 1) / TPB, TPB, 0, stream>>>(out, out, total);
    }
}